// MultiHeadAttention_Hetero2_11716670783747
// MI455X (gfx1250) — compile-verified
//
#include <hip/hip_runtime.h>
#include <hip/hip_bf16.h>

// ---------------------------------------------------------------------------
// Types for fp32 WMMA (V_WMMA_F32_16X16X4_F32):
//   A: 16x4 f32  -> 2 VGPRs/lane  (lanes 0-15: M=lane, K=k0/k0+1; lanes 16-31: K=k0+2/k0+3)
//   B: 4x16 f32  -> 2 VGPRs/lane  (lanes 0-15: N=lane, rows k0/k0+1; lanes 16-31: rows k0+2/k0+3)
//   C/D: 16x16 f32 -> 8 VGPRs/lane (lanes 0-15: M=r, N=lane; lanes 16-31: M=8+r, N=lane-16)
// ---------------------------------------------------------------------------
typedef __attribute__((ext_vector_type(2))) float v2f;
typedef __attribute__((ext_vector_type(8))) float v8f;

__device__ __forceinline__ v8f wmma_f32_k4(v2f a, v2f b, v8f c) {
  // 8 args: (neg_a, A, neg_b, B, c_mod, C, reuse_a, reuse_b)
  return __builtin_amdgcn_wmma_f32_16x16x4_f32(false, a, false, b, (short)0, c,
                                               false, false);
}

#define IN_F 64
#define OH_F 64
#define NHEAD 4
#define HDIM 16

// ---------------------------------------------------------------------------
// Zero-fill workspace accumulators (wv and z, laid out contiguously).
// ---------------------------------------------------------------------------
__global__ void zero_kernel(float* __restrict__ p, size_t count) {
  size_t i = (size_t)blockIdx.x * blockDim.x + threadIdx.x;
  if (i < count) p[i] = 0.0f;
}

// ---------------------------------------------------------------------------
// Fused Q/K/V/F projections: one wave per 16-row node tile, fp32 WMMA.
// All four 64x64 weight matrices staged in LDS (64 KB).
// ---------------------------------------------------------------------------
__global__ __launch_bounds__(128) void proj_kernel(
    const float* __restrict__ feats,
    const float* __restrict__ Wq, const float* __restrict__ bq,
    const float* __restrict__ Wk, const float* __restrict__ bk,
    const float* __restrict__ Wv, const float* __restrict__ bv,
    const float* __restrict__ Wf, const float* __restrict__ bf,
    float* __restrict__ q, float* __restrict__ k,
    float* __restrict__ v, float* __restrict__ f,
    int n_tiles) {
  __shared__ float sW[4][IN_F * OH_F];  // 4 * 16KB = 64KB
  for (int i = threadIdx.x; i < IN_F * OH_F; i += blockDim.x) {
    sW[0][i] = Wq[i];
    sW[1][i] = Wk[i];
    sW[2][i] = Wv[i];
    sW[3][i] = Wf[i];
  }
  __syncthreads();

  const int wave = threadIdx.x >> 5;
  const int lane = threadIdx.x & 31;
  const int tile = blockIdx.x * (blockDim.x >> 5) + wave;
  if (tile >= n_tiles) return;  // wave-uniform: EXEC stays all-ones for WMMA

  const int row0  = tile * 16;
  const int mrow  = lane & 15;
  const int khalf = (lane >> 4) * 2;  // 0 for lanes 0-15, 2 for lanes 16-31

  v8f acc[4][4];
#pragma unroll
  for (int m = 0; m < 4; ++m)
#pragma unroll
    for (int c = 0; c < 4; ++c) acc[m][c] = (v8f){};

  for (int k0 = 0; k0 < IN_F; k0 += 4) {
    const int kk = k0 + khalf;
    const float* ar = feats + (size_t)(row0 + mrow) * IN_F + kk;
    v2f a;
    a.x = ar[0];
    a.y = ar[1];
#pragma unroll
    for (int m = 0; m < 4; ++m) {
#pragma unroll
      for (int c = 0; c < 4; ++c) {
        const int ncol = c * 16 + (lane & 15);
        v2f b;
        b.x = sW[m][(kk + 0) * OH_F + ncol];
        b.y = sW[m][(kk + 1) * OH_F + ncol];
        acc[m][c] = wmma_f32_k4(a, b, acc[m][c]);
      }
    }
  }

  float* outs[4] = {q, k, v, f};
  const float* biases[4] = {bq, bk, bv, bf};
  const int colbase = lane & 15;
  const int rbase = (lane >> 4) * 8;
#pragma unroll
  for (int m = 0; m < 4; ++m) {
#pragma unroll
    for (int c = 0; c < 4; ++c) {
      const int col = c * 16 + colbase;
      const float bias = biases[m][col];
      float* op = outs[m] + (size_t)(row0 + rbase) * OH_F + col;
#pragma unroll
      for (int r = 0; r < 8; ++r) op[r * OH_F] = acc[m][c][r] + bias;
    }
  }
}

// ---------------------------------------------------------------------------
// Edge phase: 64 threads per edge (t = head*16 + d).
// dot(k[src],q[dst]) per head via 16-lane shfl_xor reduction, exp(clamp),
// then f32 atomic scatter of v[src]*score and score (L2-resident).
// ---------------------------------------------------------------------------
__global__ __launch_bounds__(256) void edge_kernel(
    const int* __restrict__ src, const int* __restrict__ dst,
    const float* __restrict__ q, const float* __restrict__ k,
    const float* __restrict__ v,
    float* __restrict__ wv, float* __restrict__ z, int nE) {
  const long long gid = (long long)blockIdx.x * blockDim.x + threadIdx.x;
  const long long e = gid >> 6;
  if (e >= nE) return;
  const int t = (int)(gid & 63);  // t = h*16 + d

  const int s = src[e];
  const int d = dst[e];

  const float kk = k[(size_t)s * OH_F + t];
  const float qq = q[(size_t)d * OH_F + t];
  float p = kk * qq;
  // reduce within 16-lane head groups (wave32: groups never straddle a wave)
#pragma unroll
  for (int off = 8; off >= 1; off >>= 1) p += __shfl_xor(p, off, 16);

  const float sc = expf(fminf(fmaxf(p * 0.125f, -5.0f), 5.0f));  // /sqrt(64)

  const float vv = v[(size_t)s * OH_F + t];
  atomicAdd(&wv[(size_t)d * OH_F + t], vv * sc);
  if ((t & 15) == 0) atomicAdd(&z[(size_t)d * NHEAD + (t >> 4)], sc);
}

// ---------------------------------------------------------------------------
// o = (wv / (z+1)) @ Wo + bo  — fp32 WMMA; normalization fused into A-frag.
// ---------------------------------------------------------------------------
__global__ __launch_bounds__(128) void attn_out_kernel(
    const float* __restrict__ wv, const float* __restrict__ z,
    const float* __restrict__ Wo, const float* __restrict__ bo,
    float* __restrict__ o, int n_tiles) {
  __shared__ float sW[OH_F * OH_F];  // 16KB
  for (int i = threadIdx.x; i < OH_F * OH_F; i += blockDim.x) sW[i] = Wo[i];
  __syncthreads();

  const int wave = threadIdx.x >> 5;
  const int lane = threadIdx.x & 31;
  const int tile = blockIdx.x * (blockDim.x >> 5) + wave;
  if (tile >= n_tiles) return;

  const int row0  = tile * 16;
  const int mrow  = lane & 15;
  const int khalf = (lane >> 4) * 2;

  v8f acc[4];
#pragma unroll
  for (int c = 0; c < 4; ++c) acc[c] = (v8f){};

  for (int k0 = 0; k0 < OH_F; k0 += 4) {
    const int kk = k0 + khalf;               // kk and kk+1 share a head
    const int head = kk >> 4;
    const float zr = z[(size_t)(row0 + mrow) * NHEAD + head] + 1.0f;
    const float inv = 1.0f / zr;
    const float* ar = wv + (size_t)(row0 + mrow) * OH_F + kk;
    v2f a;
    a.x = ar[0] * inv;
    a.y = ar[1] * inv;
#pragma unroll
    for (int c = 0; c < 4; ++c) {
      const int ncol = c * 16 + (lane & 15);
      v2f b;
      b.x = sW[(kk + 0) * OH_F + ncol];
      b.y = sW[(kk + 1) * OH_F + ncol];
      acc[c] = wmma_f32_k4(a, b, acc[c]);
    }
  }

  const int colbase = lane & 15;
  const int rbase = (lane >> 4) * 8;
#pragma unroll
  for (int c = 0; c < 4; ++c) {
    const int col = c * 16 + colbase;
    const float bias = bo[col];
    float* op = o + (size_t)(row0 + rbase) * OH_F + col;
#pragma unroll
    for (int r = 0; r < 8; ++r) op[r * OH_F] = acc[c][r] + bias;
  }
}

// ---------------------------------------------------------------------------
// Final: h = f + o;  out = h + LayerNorm(h).  One wave per node, 2 cols/lane.
// ---------------------------------------------------------------------------
__global__ __launch_bounds__(256) void final_kernel(
    const float* __restrict__ f, const float* __restrict__ o,
    const float* __restrict__ gamma, const float* __restrict__ beta,
    float* __restrict__ out, int n) {
  const int wave = threadIdx.x >> 5;
  const int lane = threadIdx.x & 31;
  const int node = blockIdx.x * (blockDim.x >> 5) + wave;
  if (node >= n) return;

  const size_t base = (size_t)node * OH_F;
  const float h0 = f[base + lane] + o[base + lane];
  const float h1 = f[base + 32 + lane] + o[base + 32 + lane];

  float s = h0 + h1;
#pragma unroll
  for (int off = 16; off >= 1; off >>= 1) s += __shfl_xor(s, off, 32);
  const float mu = s * (1.0f / OH_F);

  const float d0 = h0 - mu, d1 = h1 - mu;
  float vs = d0 * d0 + d1 * d1;
#pragma unroll
  for (int off = 16; off >= 1; off >>= 1) vs += __shfl_xor(vs, off, 32);
  const float rstd = rsqrtf(vs * (1.0f / OH_F) + 1e-5f);

  const float ln0 = d0 * rstd * gamma[lane] + beta[lane];
  const float ln1 = d1 * rstd * gamma[32 + lane] + beta[32 + lane];
  out[base + lane] = h0 + ln0;
  out[base + 32 + lane] = h1 + ln1;
}

// ---------------------------------------------------------------------------
// Launch
// ---------------------------------------------------------------------------
extern "C" void kernel_launch(void* const* d_in, const int* in_sizes, int n_in,
                              void* d_out, int out_size, void* d_ws, size_t ws_size,
                              hipStream_t stream) {
  const float* feats = (const float*)d_in[0];
  const float* Wq    = (const float*)d_in[1];
  const float* bq    = (const float*)d_in[2];
  const float* Wk    = (const float*)d_in[3];
  const float* bk    = (const float*)d_in[4];
  const float* Wv    = (const float*)d_in[5];
  const float* bv    = (const float*)d_in[6];
  const float* Wo    = (const float*)d_in[7];
  const float* bo    = (const float*)d_in[8];
  const float* Wf    = (const float*)d_in[9];
  const float* bf    = (const float*)d_in[10];
  const float* gamma = (const float*)d_in[11];
  const float* beta  = (const float*)d_in[12];
  const int*   src   = (const int*)d_in[13];
  const int*   dst   = (const int*)d_in[14];

  const int N = in_sizes[0] / IN_F;   // 50000
  const int E = in_sizes[13];         // 800000
  const size_t NOH = (size_t)N * OH_F;

  float* ws   = (float*)d_ws;
  float* q    = ws;                // N*64
  float* kbuf = ws + NOH;          // N*64 (reused as `o` after edge phase)
  float* vbuf = ws + 2 * NOH;      // N*64
  float* fbuf = ws + 3 * NOH;      // N*64
  float* wv   = ws + 4 * NOH;      // N*64 accumulators
  float* z    = ws + 5 * NOH;      // N*4  accumulators (contiguous with wv)
  float* o    = kbuf;              // alias: k is dead after edge_kernel

  // 1) zero the scatter accumulators (wv ++ z)
  const size_t zc = NOH + (size_t)N * NHEAD;
  zero_kernel<<<(unsigned)((zc + 255) / 256), 256, 0, stream>>>(wv, zc);

  // 2) fused Q/K/V/F projections (fp32 WMMA)
  const int n_tiles = N / 16;      // 3125 exact tiles
  proj_kernel<<<(n_tiles + 3) / 4, 128, 0, stream>>>(
      feats, Wq, bq, Wk, bk, Wv, bv, Wf, bf, q, kbuf, vbuf, fbuf, n_tiles);

  // 3) per-edge attention + scatter (L2-resident atomics)
  const long long eth = (long long)E * 64;
  edge_kernel<<<(unsigned)((eth + 255) / 256), 256, 0, stream>>>(
      src, dst, q, kbuf, vbuf, wv, z, E);

  // 4) o = (wv/(z+1)) @ Wo + bo  (fp32 WMMA, normalization fused)
  attn_out_kernel<<<(n_tiles + 3) / 4, 128, 0, stream>>>(wv, z, Wo, bo, o, n_tiles);

  // 5) h = f + o; out = h + LayerNorm(h)
  final_kernel<<<(N + 7) / 8, 256, 0, stream>>>(fbuf, o, gamma, beta,
                                                (float*)d_out, N);
}